// UltimusBlock_27788438405775
// MI455X (gfx1250) — compile-verified
//
#include <hip/hip_runtime.h>
#include <hip/hip_bf16.h>

typedef __attribute__((ext_vector_type(2))) float v2f;
typedef __attribute__((ext_vector_type(8))) float v8f;

#define IN_FEAT   48
#define OUT_FEAT  8
#define NW        8          // waves per block (wave32)
#define XS_STRIDE 52         // padded row stride (floats): conflict-free A-frag reads, 16B-aligned rows
#define ZS_STRIDE 10

__global__ __launch_bounds__(NW * 32)
void ultimus_kernel(const float* __restrict__ x,
                    const float* __restrict__ Wk, const float* __restrict__ bk,
                    const float* __restrict__ Wq, const float* __restrict__ bq,
                    const float* __restrict__ Wv, const float* __restrict__ bv,
                    const float* __restrict__ Wfc, const float* __restrict__ bfc,
                    float* __restrict__ out, int ntiles)
{
    // Per-wave private LDS slices (DS ops are in-order within a wave; no cross-wave sharing)
    __shared__ __align__(16) float xs_all[NW][16 * XS_STRIDE];
    __shared__ __align__(16) float kq_all[NW][16 * 16];
    __shared__ __align__(16) float vv_all[NW][16 * 8];
    __shared__ __align__(16) float zz_all[NW][16 * ZS_STRIDE];

    const int lane = threadIdx.x & 31;
    const int wid  = threadIdx.x >> 5;
    float* xs  = xs_all[wid];
    float* kqs = kq_all[wid];
    float* vvs = vv_all[wid];
    float* zzs = zz_all[wid];

    const int col  = lane & 15;   // N index of this lane in B/C fragments
    const int half = lane >> 4;   // upper/lower half-wave

    // ---- register-resident weight fragments (B operands of v_wmma_f32_16x16x4_f32) ----
    // B layout: VGPR j, lane l -> K = j + 2*(l>>4), N = l&15   (4x16 tile, 2 VGPRs)
    v2f Bkq[12], Bv[12], Bfc[3][2];
#pragma unroll
    for (int k = 0; k < 12; ++k) {
#pragma unroll
        for (int j = 0; j < 2; ++j) {
            const int c = 4 * k + j + 2 * half;                       // K index in 0..47
            Bkq[k][j] = (col < 8) ? Wk[col * IN_FEAT + c]             // cols 0-7  -> k
                                  : Wq[(col - 8) * IN_FEAT + c];      // cols 8-15 -> q
            Bv[k][j]  = (col < 8) ? Wv[col * IN_FEAT + c] : 0.f;      // cols 0-7  -> v
        }
    }
#pragma unroll
    for (int t = 0; t < 3; ++t)
#pragma unroll
        for (int kc = 0; kc < 2; ++kc)
#pragma unroll
            for (int j = 0; j < 2; ++j) {
                const int kk = 4 * kc + j + 2 * half;                 // K in 0..7
                Bfc[t][kc][j] = Wfc[(16 * t + col) * OUT_FEAT + kk];  // WfcT[kk][n], n = 16t+col
            }

    const float bkq_lane = (col < 8) ? bk[col] : bq[col - 8];
    const float bv_lane  = (col < 8) ? bv[col] : 0.f;
    float bfc_lane[3];
#pragma unroll
    for (int t = 0; t < 3; ++t) bfc_lane[t] = bfc[16 * t + col];

    const int gw     = blockIdx.x * NW + wid;
    const int nwaves = gridDim.x * NW;
    const size_t tile_stride = (size_t)nwaves * (16 * IN_FEAT);

    // per-lane global base pointers; advance by constant stride -> immediate-offset vmem ops
    const float* xg = x   + (size_t)gw * (16 * IN_FEAT) + lane * 4;           // contiguous float4 slices
    float*       og = out + (size_t)gw * (16 * IN_FEAT) + half * 8 * IN_FEAT + col;

    // per-lane LDS staging addresses (loop-invariant)
    float* xls[6];
#pragma unroll
    for (int i = 0; i < 6; ++i) {
        const int idx = i * 32 + lane;                 // float4 index, 192 per tile
        xls[i] = xs + (idx / 12) * XS_STRIDE + (idx % 12) * 4;
    }
    const float* xfrag = xs + (lane & 15) * XS_STRIDE + 2 * half;   // A-frag base (k-chunk adds 4k)
    const float* zfrag = zzs + (lane & 15) * ZS_STRIDE + 2 * half;

    for (int tile = gw; tile < ntiles; tile += nwaves) {
        // hint-prefetch next tile (global_prefetch_b8)
        if (tile + nwaves < ntiles)
            __builtin_prefetch(xg + tile_stride, 0, 3);

        __builtin_amdgcn_wave_barrier();
        // ---- coalesced 16x48 f32 tile load -> LDS (global side is purely contiguous) ----
#pragma unroll
        for (int i = 0; i < 6; ++i) {
            const float4 val = *(const float4*)(xg + i * 128);      // 512B immediate offsets
            *(float4*)xls[i] = val;
        }
        __builtin_amdgcn_wave_barrier();

        // ---- projections: X(16x48) @ [Wk|Wq]T(48x16) and @ WvT(48x8), K in 12 steps of 4 ----
        v8f accP = {};   // k|q tile
        v8f accV = {};   // v tile (cols 0-7 valid)
#pragma unroll
        for (int k = 0; k < 12; ++k) {
            // A layout: lane m(0-15): V0=K0,V1=K1 ; lane m+16: V0=K2,V1=K3
            v2f a = *(const v2f*)(xfrag + 4 * k);
            accP = __builtin_amdgcn_wmma_f32_16x16x4_f32(false, a, false, Bkq[k], (short)0, accP, false, false);
            accV = __builtin_amdgcn_wmma_f32_16x16x4_f32(false, a, false, Bv[k],  (short)0, accV, false, false);
        }

        // ---- scatter C tiles to canonical [sample][feature] LDS layout, add biases ----
        // C layout: VGPR i, lane l -> row = i + 8*(l>>4), col = l&15
#pragma unroll
        for (int i = 0; i < 8; ++i) {
            const int s = i + 8 * half;
            kqs[s * 16 + col] = accP[i] + bkq_lane;
            if (col < 8) vvs[s * 8 + col] = accV[i] + bv_lane;
        }
        __builtin_amdgcn_wave_barrier();

        // ---- per-sample softmax over rows; 2 lanes/sample, 4 output cols each ----
        {
            const int s  = lane & 15;
            const int jb = half * 4;
            const float LOG2E     = 1.4426950408889634f;
            const float inv_scale = 0.35355339059327373f;  // 1/sqrt(8)
            float ql[8], vv[8];
#pragma unroll
            for (int i = 0; i < 8; ++i) {
                ql[i] = kqs[s * 16 + 8 + i] * LOG2E;       // fold log2(e) into q once
                vv[i] = vvs[s * 8 + i];
            }
            float qmax = ql[0], qmin = ql[0];
#pragma unroll
            for (int i = 1; i < 8; ++i) { qmax = fmaxf(qmax, ql[i]); qmin = fminf(qmin, ql[i]); }
#pragma unroll
            for (int j = 0; j < 4; ++j) {
                const float kj = kqs[s * 16 + jb + j] * inv_scale;
                const float m2 = fmaxf(kj * qmax, kj * qmin);   // max_i (ql_i*kj), linear in ql
                float den = 0.f, num = 0.f;
#pragma unroll
                for (int i = 0; i < 8; ++i) {
                    const float e = exp2f(fmaf(ql[i], kj, -m2));  // v_fma + v_exp_f32
                    den += e;
                    num = fmaf(vv[i], e, num);
                }
                zzs[s * ZS_STRIDE + jb + j] = num * __builtin_amdgcn_rcpf(den);
            }
        }
        __builtin_amdgcn_wave_barrier();

        // ---- fc: Z(16x8) @ WfcT(8x48): 2 K-chunks x 3 N-tiles of WMMA ----
        v8f accO[3] = {{}, {}, {}};
#pragma unroll
        for (int kc = 0; kc < 2; ++kc) {
            v2f za = *(const v2f*)(zfrag + 4 * kc);
#pragma unroll
            for (int t = 0; t < 3; ++t)
                accO[t] = __builtin_amdgcn_wmma_f32_16x16x4_f32(false, za, false, Bfc[t][kc], (short)0, accO[t], false, false);
        }

        // ---- store 16x48 output tile (+bfc); single base, immediate offsets ----
#pragma unroll
        for (int t = 0; t < 3; ++t)
#pragma unroll
            for (int i = 0; i < 8; ++i)
                og[i * IN_FEAT + 16 * t] = accO[t][i] + bfc_lane[t];   // byte off = i*192 + t*64

        xg += tile_stride;
        og += tile_stride;
        __builtin_amdgcn_wave_barrier();
    }
}

extern "C" void kernel_launch(void* const* d_in, const int* in_sizes, int n_in,
                              void* d_out, int out_size, void* d_ws, size_t ws_size,
                              hipStream_t stream) {
    const float* x   = (const float*)d_in[0];
    const float* Wk  = (const float*)d_in[1];
    const float* bk  = (const float*)d_in[2];
    const float* Wq  = (const float*)d_in[3];
    const float* bq  = (const float*)d_in[4];
    const float* Wv  = (const float*)d_in[5];
    const float* bv  = (const float*)d_in[6];
    const float* Wfc = (const float*)d_in[7];
    const float* bfc = (const float*)d_in[8];
    float* out = (float*)d_out;

    const int Btot   = in_sizes[0] / IN_FEAT;   // 1048576
    const int ntiles = Btot / 16;               // 16 samples per wave-tile

    int blocks = 4096;                          // 8 waves/block -> 32768 waves, grid-stride over tiles
    if (blocks * NW > ntiles) blocks = (ntiles + NW - 1) / NW;

    ultimus_kernel<<<blocks, NW * 32, 0, stream>>>(x, Wk, bk, Wq, bq, Wv, bv, Wfc, bfc, out, ntiles);
}